// UpFIRDnConv2d_26525718020494
// MI455X (gfx1250) — compile-verified
//
#include <hip/hip_runtime.h>
#include <hip/hip_bf16.h>

// ---------------------------------------------------------------------------
// Kernel 1: 3x3 "same" conv 64->64 as implicit GEMM on the fp32 matrix pipe.
//   M = 64 output channels (4 tiles of 16)
//   N = 64-pixel strips of one image row (4 waves x 16 pixels)
//   K = C_IN = 64 per filter tap (9 accumulation passes), 16 WMMA K=4 steps
// Uses V_WMMA_F32_16X16X4_F32 to stay in full fp32 precision, and
// GLOBAL_LOAD_ASYNC_TO_LDS_B32 (ASYNCcnt) to stage the input tile without
// bouncing through VGPRs.
// ---------------------------------------------------------------------------

typedef __attribute__((ext_vector_type(2))) float v2f;
typedef __attribute__((ext_vector_type(8))) float v8f;

#define HW 128
#define CIN 64
#define COUT 64

// LDS layout strides (floats), padded for bank-conflict-free access:
//   Xs: per-ci stride 200 (3 rows x 66 cols used) -> half-wave K offset lands
//       16 banks apart (400 mod 64 == 16), disjoint from the m16 span.
//   Ws: per-co stride 68 -> 4*m16 + 2*half banks, halves on distinct parity.
#define XS_STRIDE 200
#define WS_STRIDE 68

__global__ __launch_bounds__(128)
void conv3x3_wmma_f32_kernel(const float* __restrict__ x,
                             const float* __restrict__ w,      // (64,64,3,3)
                             const float* __restrict__ conv_b, // (64,)
                             const float* __restrict__ bias,   // (1,64,1,1)
                             float* __restrict__ y0)           // (8,64,128,128)
{
    extern __shared__ float smem[];
    float* Xs = smem;                       // [CIN][3][66] (stride XS_STRIDE)
    float* Ws = smem + CIN * XS_STRIDE;     // [COUT][CIN]  (stride WS_STRIDE)

    const int tid  = threadIdx.x;
    const int lane = tid & 31;
    const int wave = tid >> 5;
    const int half = lane >> 4;   // K-offset selector (ISA f32 16x4 layout)
    const int m16  = lane & 15;   // M (A-frag) / N (B-frag) index

    const int wBase = blockIdx.x * 64;   // 2 strips per row
    const int h     = blockIdx.y;        // image row
    const int b     = blockIdx.z;        // batch

    // ---- stage input halo tile: rows h-1..h+1, cols wBase-1..wBase+64 ----
    // Interior elements: async global->LDS DMA (no VGPR round trip).
    // Out-of-image halo: ds_store of zero (disjoint LDS addresses).
    const float* xb = x + (size_t)b * CIN * HW * HW;
    __builtin_prefetch(xb + (size_t)0 * HW * HW + h * HW + wBase, 0, 0);
    for (int idx = tid; idx < CIN * 3 * 66; idx += 128) {
        int ci  = idx / 198;
        int rem = idx - ci * 198;
        int r   = rem / 66;
        int cc  = rem - r * 66;
        int gr  = h - 1 + r;
        int gc  = wBase - 1 + cc;
        float* dst = &Xs[ci * XS_STRIDE + r * 66 + cc];
        if (gr >= 0 && gr < HW && gc >= 0 && gc < HW) {
            // LDS byte offset = low 32 bits of the flat shared address
            // (aperture rule).
            unsigned ldsoff = (unsigned)(uintptr_t)dst;
            const float* gptr = xb + (size_t)ci * HW * HW + gr * HW + gc;
            asm volatile("global_load_async_to_lds_b32 %0, %1, off"
                         :: "v"(ldsoff), "v"(gptr)
                         : "memory");
        } else {
            *dst = 0.0f;
        }
    }
    // Drain the async copies before anyone reads the tile.
    asm volatile("s_wait_asynccnt 0" ::: "memory");

    v8f acc[4];
#pragma unroll
    for (int t = 0; t < 4; ++t) acc[t] = (v8f){0, 0, 0, 0, 0, 0, 0, 0};

    const int pl = wave * 16 + m16;      // this lane's pixel within the strip

    for (int khkw = 0; khkw < 9; ++khkw) {
        const int kh = khkw / 3;
        const int kw = khkw - kh * 3;

        __syncthreads();
        // stage W[., ., kh, kw] as a 64x64 GEMM A-matrix
        for (int idx = tid; idx < COUT * CIN; idx += 128) {
            int co = idx >> 6;
            int ci = idx & 63;
            Ws[co * WS_STRIDE + ci] = w[((co * CIN + ci) * 3 + kh) * 3 + kw];
        }
        __syncthreads();

        const float* xrow = &Xs[kh * 66 + kw];   // + ci*XS_STRIDE + pl
#pragma unroll
        for (int kc = 0; kc < 16; ++kc) {
            const int ciA = kc * 4 + half * 2;   // this half-wave's K pair
            v2f bf;                               // B-frag: x[ci][pixel]
            bf.x = xrow[ciA * XS_STRIDE + pl];
            bf.y = xrow[(ciA + 1) * XS_STRIDE + pl];
#pragma unroll
            for (int t = 0; t < 4; ++t) {
                v2f af;                           // A-frag: w[co][ci]
                af.x = Ws[(t * 16 + m16) * WS_STRIDE + ciA];
                af.y = Ws[(t * 16 + m16) * WS_STRIDE + ciA + 1];
                acc[t] = __builtin_amdgcn_wmma_f32_16x16x4_f32(
                    false, af, false, bf, (short)0, acc[t], false, false);
            }
        }
    }

    // ---- store: acc[t][j] -> co = t*16 + 8*half + j, col = wBase + pl ----
    const int wcol = wBase + pl;
#pragma unroll
    for (int t = 0; t < 4; ++t) {
#pragma unroll
        for (int j = 0; j < 8; ++j) {
            int co  = t * 16 + half * 8 + j;
            float v = acc[t][j] + conv_b[co] + bias[co];
            y0[(((size_t)b * COUT + co) * HW + h) * HW + wcol] = v;
        }
    }
}

// ---------------------------------------------------------------------------
// Kernel 2: fused  up(x2, nearest) -> 12x12 FIR -> lrelu*sqrt(2)
//                 -> 12x12 FIR -> down(::2)
// Separable (filters are outer products; 1D factors recovered on device).
// One 16x16 output tile per block; the 256x256 intermediate lives in LDS only.
// ---------------------------------------------------------------------------
__global__ __launch_bounds__(256)
void upfirdn_lrelu_kernel(const float* __restrict__ y0,   // (512,128,128)
                          const float* __restrict__ upf,  // (12,12)
                          const float* __restrict__ dnf,  // (12,12)
                          float* __restrict__ out)        // (512,128,128)
{
    __shared__ float u1[12], d1[12];
    __shared__ float s0[27 * 28];   // y0 patch rows i0-5..i0+21, cols j0-5..j0+21
    __shared__ float Hr[27 * 44];   // horizontal up-FIR at y0-row resolution
    __shared__ float t1[42 * 44];   // lrelu'd up-FIR tile (42x42 of y1)
    __shared__ float Dh[42 * 16];   // horizontal down-FIR, decimated cols

    const int tid = threadIdx.x;
    const int j0  = blockIdx.x * 16;
    const int i0  = blockIdx.y * 16;
    const int bc  = blockIdx.z;                     // b*64 + c
    const float* src = y0 + (size_t)bc * HW * HW;

    // Recover 1D factors: F = outer(f,f)  =>  f[k] = F[k][5] / sqrt(F[5][5])
    // (sign-consistent up to a global flip, which cancels in the product).
    if (tid < 12)
        u1[tid] = upf[tid * 12 + 5] * rsqrtf(upf[5 * 12 + 5]);
    else if (tid < 24)
        d1[tid - 12] = dnf[(tid - 12) * 12 + 5] * rsqrtf(dnf[5 * 12 + 5]);

    __builtin_prefetch(src + (size_t)(i0 > 5 ? i0 - 5 : 0) * HW + j0, 0, 0);

    // ---- stage y0 patch (zero-padded at image borders) ----
    for (int idx = tid; idx < 27 * 27; idx += 256) {
        int r  = idx / 27, c = idx - r * 27;
        int gr = i0 - 5 + r, gc = j0 - 5 + c;
        float v = 0.0f;
        if (gr >= 0 && gr < HW && gc >= 0 && gc < HW)
            v = src[gr * HW + gc];
        s0[r * 28 + c] = v;
    }
    __syncthreads();

    // ---- horizontal up-FIR: Hr[r][vi], v = 2*j0-5+vi  (repeat-upsampled) --
    for (int idx = tid; idx < 27 * 42; idx += 256) {
        int r = idx / 42, vi = idx - r * 42;
        int v = 2 * j0 - 5 + vi;
        float acc = 0.0f;
#pragma unroll
        for (int l = 0; l < 12; ++l) {
            int p = v - 5 + l;                    // upsampled column
            if (p >= 0 && p < 2 * HW)
                acc += u1[l] * s0[r * 28 + ((p >> 1) - (j0 - 5))];
        }
        Hr[r * 44 + vi] = acc;
    }
    __syncthreads();

    // ---- vertical up-FIR + leaky-ReLU*gain: t1[ui][vi] (42x42 of y1) ----
    const float GAINF = 1.41421356237f;
    for (int idx = tid; idx < 42 * 42; idx += 256) {
        int ui = idx / 42, vi = idx - ui * 42;
        int u = 2 * i0 - 5 + ui;
        int v = 2 * j0 - 5 + vi;
        float r = 0.0f;
        if (u >= 0 && u < 2 * HW && v >= 0 && v < 2 * HW) {
            float acc = 0.0f;
#pragma unroll
            for (int k = 0; k < 12; ++k) {
                int p = u - 5 + k;                // upsampled row
                if (p >= 0 && p < 2 * HW)
                    acc += u1[k] * Hr[((p >> 1) - (i0 - 5)) * 44 + vi];
            }
            r = (acc >= 0.0f ? acc : 0.2f * acc) * GAINF;
        }
        t1[ui * 44 + vi] = r;                     // zero outside [0,256)^2
    }
    __syncthreads();

    // ---- horizontal down-FIR at decimated columns: Dh[ui][jj] ----
    for (int idx = tid; idx < 42 * 16; idx += 256) {
        int ui = idx / 16, jj = idx - ui * 16;
        float acc = 0.0f;
#pragma unroll
        for (int b2 = 0; b2 < 12; ++b2)
            acc += d1[b2] * t1[ui * 44 + 2 * jj + b2];
        Dh[ui * 16 + jj] = acc;
    }
    __syncthreads();

    // ---- vertical down-FIR + decimated store: one output pixel/thread ----
    {
        int ii = tid >> 4, jj = tid & 15;
        float acc = 0.0f;
#pragma unroll
        for (int a = 0; a < 12; ++a)
            acc += d1[a] * Dh[(2 * ii + a) * 16 + jj];
        out[((size_t)bc * HW + (i0 + ii)) * HW + (j0 + jj)] = acc;
    }
}

// ---------------------------------------------------------------------------
// Launch
// ---------------------------------------------------------------------------
extern "C" void kernel_launch(void* const* d_in, const int* in_sizes, int n_in,
                              void* d_out, int out_size, void* d_ws, size_t ws_size,
                              hipStream_t stream) {
    (void)in_sizes; (void)n_in; (void)out_size; (void)ws_size;
    const float* x      = (const float*)d_in[0];
    const float* conv_w = (const float*)d_in[1];
    const float* conv_b = (const float*)d_in[2];
    const float* bias   = (const float*)d_in[3];
    const float* upf    = (const float*)d_in[4];
    const float* dnf    = (const float*)d_in[5];
    float* y0  = (float*)d_ws;     // 8*64*128*128 fp32 = 33.5 MB scratch
    float* out = (float*)d_out;

    // Kernel 1: one 64-channel x 64-pixel strip per block (4 waves).
    dim3 g1(HW / 64, HW, 8);
    size_t smem1 = (size_t)(CIN * XS_STRIDE + COUT * WS_STRIDE) * sizeof(float);
    conv3x3_wmma_f32_kernel<<<g1, 128, smem1, stream>>>(x, conv_w, conv_b, bias, y0);

    // Kernel 2: one 16x16 output tile per block, 512 channel-images.
    dim3 g2(HW / 16, HW / 16, 8 * 64);
    upfirdn_lrelu_kernel<<<g2, 256, 0, stream>>>(y0, upf, dnf, out);
}